// MultiLabelEmbedding_88184268521790
// MI455X (gfx1250) — compile-verified
//
#include <hip/hip_runtime.h>

// MultiLabelEmbedding (EmbeddingBag mean with ragged lengths) for gfx1250.
//
// One wave32 per bag; lane d owns output dim d (D=32 == wave32).
// Per 4-index chunk:
//   A (16x4 f32)  = mask (all-ones for full chunks) broadcast to all 16 rows
//   B (4x16 f32)  = 4 gathered embedding rows, one 16-dim tile per WMMA
//   C (16x16 f32) accumulates; all C rows identical == bag sum.
// Index/length traffic goes through the scalar cache (bag id forced uniform
// via readfirstlane -> s_load_b128 per chunk), table gathers are coalesced
// 64B vector loads in saddr+voffset form (32-bit byte offsets: idx*128+n*4
// < 12.8MB), reduction runs on the XDL pipe (v_wmma f32 => RNE f32 MACs,
// numerically identical to the f32 reference).

typedef __attribute__((ext_vector_type(2))) float v2f;
typedef __attribute__((ext_vector_type(8))) float v8f;

#define BAG_DIM         32
#define WAVES_PER_BLOCK 8

__global__ __launch_bounds__(256)
void MultiLabelEmbedding_88184268521790_kernel(
    const float* __restrict__ table,    // [NUM_LABELS, 32]
    const int*   __restrict__ indices,  // [nBags, L]
    const int*   __restrict__ lengths,  // [nBags]
    float*       __restrict__ out,      // [nBags, 32]
    int nBags, int L)
{
    const int lane = threadIdx.x & 31;
    const int wave = threadIdx.x >> 5;
    // Wave-uniform by construction; readfirstlane tells the compiler, so the
    // length/index loads lower to SMEM (s_load) and the loop runs on SALU.
    const int bag = __builtin_amdgcn_readfirstlane(blockIdx.x * WAVES_PER_BLOCK + wave);
    if (bag >= nBags) return;

    const int len = lengths[bag];                       // s_load (uniform)
    const int* __restrict__ bi = indices + (long)bag * L;
    const char* __restrict__ tb = (const char*)table;   // uniform byte base

    const int      half = lane >> 4;   // which K-pair this half-wave supplies
    const unsigned nb   = (unsigned)((lane & 15) * 4);  // column byte offset
    const bool     hi   = (lane >= 16);

    v8f c0 = {};  // dims  0..15 accumulator (16x16 f32 C)
    v8f c1 = {};  // dims 16..31 accumulator
    const v2f aOne = {1.0f, 1.0f};

    // ---- full chunks: no masking, consecutive scalar index loads ----
    const int kfull = len & ~3;
#pragma unroll 2
    for (int k0 = 0; k0 < kfull; k0 += 4) {
        const int i0 = bi[k0 + 0];                      // s_load_b128
        const int i1 = bi[k0 + 1];
        const int i2 = bi[k0 + 2];
        const int i3 = bi[k0 + 3];
        // 32-bit byte offsets -> saddr + voffset global loads
        const unsigned o0 = ((unsigned)(hi ? i2 : i0) << 7) + nb;  // K slot VGPR0
        const unsigned o1 = ((unsigned)(hi ? i3 : i1) << 7) + nb;  // K slot VGPR1
        v2f b0, b1;
        b0.x = *(const float*)(tb + o0);                // dims  0..15
        b0.y = *(const float*)(tb + o1);
        b1.x = *(const float*)(tb + o0 + 64);           // dims 16..31
        b1.y = *(const float*)(tb + o1 + 64);
        c0 = __builtin_amdgcn_wmma_f32_16x16x4_f32(
                 false, aOne, false, b0, (short)0, c0, false, false);
        c1 = __builtin_amdgcn_wmma_f32_16x16x4_f32(
                 false, aOne, false, b1, (short)0, c1, false, false);
    }

    // ---- tail chunk: uniform clamped slots keep index loads scalar; the
    //      A-mask zeroes the padded K slots ----
    if (kfull < len) {
        const int k0   = kfull;
        const int last = len - 1;                       // len >= 1 here
        const int t0 = bi[(k0 + 0 <= last) ? k0 + 0 : last];
        const int t1 = bi[(k0 + 1 <= last) ? k0 + 1 : last];
        const int t2 = bi[(k0 + 2 <= last) ? k0 + 2 : last];
        const int t3 = bi[(k0 + 3 <= last) ? k0 + 3 : last];
        const unsigned o0 = ((unsigned)(hi ? t2 : t0) << 7) + nb;
        const unsigned o1 = ((unsigned)(hi ? t3 : t1) << 7) + nb;
        const int j0 = k0 + 2 * half;
        const int j1 = j0 + 1;
        v2f a;
        a.x = (j0 < len) ? 1.0f : 0.0f;
        a.y = (j1 < len) ? 1.0f : 0.0f;
        v2f b0, b1;
        b0.x = *(const float*)(tb + o0);
        b0.y = *(const float*)(tb + o1);
        b1.x = *(const float*)(tb + o0 + 64);
        b1.y = *(const float*)(tb + o1 + 64);
        c0 = __builtin_amdgcn_wmma_f32_16x16x4_f32(
                 false, a, false, b0, (short)0, c0, false, false);
        c1 = __builtin_amdgcn_wmma_f32_16x16x4_f32(
                 false, a, false, b1, (short)0, c1, false, false);
    }

    // All C rows are identical; VGPR0 holds column n for this lane in both
    // halves, so lane d (0..31) owns output dim d directly.
    const float denom = (float)((len > 0) ? len : 1);
    const float val   = ((lane < 16) ? c0[0] : c1[0]) / denom;
    __builtin_nontemporal_store(val, &out[(size_t)bag * BAG_DIM + lane]);
}

extern "C" void kernel_launch(void* const* d_in, const int* in_sizes, int n_in,
                              void* d_out, int out_size, void* d_ws, size_t ws_size,
                              hipStream_t stream) {
    const float* table   = (const float*)d_in[0];
    const int*   indices = (const int*)d_in[1];
    const int*   lengths = (const int*)d_in[2];
    float*       out     = (float*)d_out;

    const int nBags = in_sizes[2];               // 16384
    const int L     = in_sizes[1] / nBags;       // 50

    const int blocks = (nBags + WAVES_PER_BLOCK - 1) / WAVES_PER_BLOCK;
    MultiLabelEmbedding_88184268521790_kernel<<<blocks, 256, 0, stream>>>(
        table, indices, lengths, out, nBags, L);
}